// DilateLoss_52613349376644
// MI455X (gfx1250) — compile-verified
//
#include <hip/hip_runtime.h>
#include <hip/hip_bf16.h>

typedef __attribute__((ext_vector_type(2))) float v2f;
typedef __attribute__((ext_vector_type(8))) float v8f;

#define DN 256
#define DB 128
#define BIGF 1e8f
#define GAMMA_F 0.01f
#define INVG 100.0f

// ---------------------------------------------------------------------------
// Kernel 1: cost matrix D[b,i,j] = (t_i - p_j)^2 built tile-wise with
// V_WMMA_F32_16X16X4_F32:  D_tile = A(t) x B(-2p) + C(t^2 + p^2).
// One wave (32 lanes) per (batch, tile-row); loops over 16 tile-columns.
// ---------------------------------------------------------------------------
__global__ __launch_bounds__(32) void dilate_cost_wmma(
    const float* __restrict__ pred, const float* __restrict__ target,
    float* __restrict__ Dw) {
  const int b = blockIdx.x;
  const int it = blockIdx.y;           // tile row 0..15
  const int lane = threadIdx.x;        // 0..31
  const int half = lane >> 4;
  const int l = lane & 15;
  const int i0 = it * 16;
  const float* t = target + b * DN;
  const float* p = pred + b * DN;

  // C-layout: VGPR r holds M = r + 8*half, N = l
  float t8[8];
#pragma unroll
  for (int r = 0; r < 8; ++r) t8[r] = t[i0 + r + 8 * half];

  // A (16x4 f32): lanes 0-15 hold M=0..15 with VGPR0=K0, VGPR1=K1;
  // lanes 16-31 hold K2,K3 (zeroed -> rank-1 product over K=0 only).
  v2f a;
  a.x = (half == 0) ? t[i0 + l] : 0.0f;
  a.y = 0.0f;

  float* Drow = Dw + (size_t)b * DN * DN;
  for (int jt = 0; jt < 16; ++jt) {
    const int j0 = jt * 16;
    const float pj = p[j0 + l];        // N = l for both halves
    v2f bb;
    bb.x = (half == 0) ? (-2.0f * pj) : 0.0f;  // B row K=0
    bb.y = 0.0f;
    v8f c;
#pragma unroll
    for (int r = 0; r < 8; ++r) c[r] = t8[r] * t8[r] + pj * pj;
    c = __builtin_amdgcn_wmma_f32_16x16x4_f32(
        /*neg_a=*/false, a, /*neg_b=*/false, bb,
        /*c_mod=*/(short)0, c, /*reuse_a=*/false, /*reuse_b=*/false);
#pragma unroll
    for (int r = 0; r < 8; ++r) {
      const int M = r + 8 * half;
      Drow[(i0 + M) * DN + (j0 + l)] = c[r];
    }
  }
}

// ---------------------------------------------------------------------------
// Kernel 2: per-batch soft-DTW forward + analytic backward (gradient path),
// anti-diagonal wavefront, one 256-thread workgroup per batch element.
// R neighbor values and E values live in rotating LDS diagonal buffers.
// R is materialized to workspace for the backward replay (fits in L2).
// ---------------------------------------------------------------------------
__global__ __launch_bounds__(256) void dilate_dp(
    const float* __restrict__ Dw, float* __restrict__ Rg,
    float* __restrict__ pshape, float* __restrict__ ptemp) {
  __shared__ float Rbuf[3][260];
  __shared__ float Ebuf[3][264];
  __shared__ float red[256];

  const int b = blockIdx.x;
  const int tid = threadIdx.x;
  const int j = tid + 1;  // column 1..256
  const float* Db = Dw + (size_t)b * DN * DN;
  float* Rb = Rg + (size_t)b * DN * DN;

  for (int k = tid; k < 3 * 260; k += 256) ((float*)Rbuf)[k] = BIGF;
  for (int k = tid; k < 3 * 264; k += 256) ((float*)Ebuf)[k] = 0.0f;
  __syncthreads();
  if (tid == 0) Rbuf[0][0] = 0.0f;  // R[0][0]=0; all other borders BIG
  __syncthreads();

  // ---------------- forward: R[i,j] = D + softmin(diag, up, left) ---------
  for (int d = 2; d <= 2 * DN; ++d) {
    const int i = d - j;
    float* cur = Rbuf[d % 3];
    const float* pm1 = Rbuf[(d + 2) % 3];  // diagonal d-1
    const float* pm2 = Rbuf[(d + 1) % 3];  // diagonal d-2
    if (i >= 1 && i <= DN) {
      const float diag = pm2[j - 1];
      const float up = pm1[j];
      const float left = pm1[j - 1];
      const float dij = Db[(i - 1) * DN + (j - 1)];
      const float m = fminf(diag, fminf(up, left));
      const float s = __expf((m - diag) * INVG) + __expf((m - up) * INVG) +
                      __expf((m - left) * INVG);
      const float r = dij + m - GAMMA_F * __logf(s);
      cur[j] = r;
      Rb[(i - 1) * DN + (j - 1)] = r;
    } else if (i == 0) {
      cur[j] = BIGF;  // border R[0][d]
    }
    if (tid == 0) cur[0] = BIGF;  // border R[d][0]
    __syncthreads();
  }
  if (tid == 0) pshape[b] = Rbuf[(2 * DN) % 3][DN];  // R[N][N]

  // ---------------- backward: E[i,j] = sum_succ E[s]*exp((R[s]-D[s]-R)/g) -
  __syncthreads();
  for (int k = tid; k < 3 * 260; k += 256) ((float*)Rbuf)[k] = BIGF;
  for (int k = tid; k < 3 * 264; k += 256) ((float*)Ebuf)[k] = 0.0f;
  __syncthreads();
  if (tid == 0) {
    Rbuf[(2 * DN) % 3][DN] = Rb[(DN - 1) * DN + (DN - 1)];
    Ebuf[(2 * DN) % 3][DN] = 1.0f;  // E[N][N] = 1
  }
  __syncthreads();

  float S = 0.0f;  // per-thread sum of E[i,j]*(i-j)^2
  for (int d = 2 * DN - 1; d >= 2; --d) {
    const int i = d - j;
    float* curR = Rbuf[d % 3];
    float* curE = Ebuf[d % 3];
    const float* R1 = Rbuf[(d + 1) % 3];
    const float* R2 = Rbuf[(d + 2) % 3];
    const float* E1 = Ebuf[(d + 1) % 3];
    const float* E2 = Ebuf[(d + 2) % 3];
    if (i >= 1 && i <= DN) {
      const float rij = Rb[(i - 1) * DN + (j - 1)];
      float acc = 0.0f;
      if (i + 1 <= DN) {  // successor (i+1, j) on diag d+1 at position j
        acc += E1[j] * __expf((R1[j] - Db[i * DN + (j - 1)] - rij) * INVG);
      }
      if (j + 1 <= DN) {  // successor (i, j+1) on diag d+1 at position j+1
        acc += E1[j + 1] * __expf((R1[j + 1] - Db[(i - 1) * DN + j] - rij) * INVG);
      }
      if (i + 1 <= DN && j + 1 <= DN) {  // successor (i+1, j+1) on diag d+2
        acc += E2[j + 1] * __expf((R2[j + 1] - Db[i * DN + j] - rij) * INVG);
      }
      curR[j] = rij;
      curE[j] = acc;
      const float dd = (float)(i - j);
      S += acc * dd * dd;
    }
    __syncthreads();
  }

  red[tid] = S;
  __syncthreads();
  for (int off = 128; off > 0; off >>= 1) {
    if (tid < off) red[tid] += red[tid + off];
    __syncthreads();
  }
  if (tid == 0) ptemp[b] = red[0];
}

// ---------------------------------------------------------------------------
// Kernel 3: deterministic final reduction to the scalar loss.
// ---------------------------------------------------------------------------
__global__ __launch_bounds__(128) void dilate_finalize(
    const float* __restrict__ pshape, const float* __restrict__ ptemp,
    float* __restrict__ out) {
  __shared__ float s1[128];
  __shared__ float s2[128];
  const int t = threadIdx.x;
  s1[t] = pshape[t];
  s2[t] = ptemp[t];
  __syncthreads();
  for (int off = 64; off > 0; off >>= 1) {
    if (t < off) {
      s1[t] += s1[t + off];
      s2[t] += s2[t + off];
    }
    __syncthreads();
  }
  if (t == 0) {
    const float loss_shape = s1[0] / (float)DB;
    const float loss_temporal = s2[0] / ((float)DB * (float)DN * (float)DN);
    out[0] = 0.8f * loss_shape + 0.2f * loss_temporal;
  }
}

// ---------------------------------------------------------------------------
extern "C" void kernel_launch(void* const* d_in, const int* in_sizes, int n_in,
                              void* d_out, int out_size, void* d_ws,
                              size_t ws_size, hipStream_t stream) {
  const float* pred = (const float*)d_in[0];    // (128,256,1)
  const float* target = (const float*)d_in[1];  // (128,256,1)
  float* out = (float*)d_out;                   // scalar
  float* ws = (float*)d_ws;

  // workspace layout (floats)
  float* Dw = ws;                                  // 128*256*256
  float* Rg = ws + (size_t)DB * DN * DN;           // 128*256*256
  float* pshape = ws + 2 * (size_t)DB * DN * DN;   // 128
  float* ptemp = pshape + DB;                      // 128

  dilate_cost_wmma<<<dim3(DB, 16), 32, 0, stream>>>(pred, target, Dw);
  dilate_dp<<<DB, 256, 0, stream>>>(Dw, Rg, pshape, ptemp);
  dilate_finalize<<<1, 128, 0, stream>>>(pshape, ptemp, out);
}